// NeighborIntegration_54305566491422
// MI455X (gfx1250) — compile-verified
//
#include <hip/hip_runtime.h>
#include <hip/hip_bf16.h>

// Problem constants (fixed by the reference)
#define BATCH 4096
#define NN    64      // neighbors
#define DD    128     // embedding dim
#define LROW  132     // padded LDS row stride (floats) for k1
#define ASTR  260     // padded LDS row stride (floats) for k2 A-tile

typedef __attribute__((ext_vector_type(2))) float v2f;
typedef __attribute__((ext_vector_type(8))) float v8f;

// ---------------------------------------------------------------------------
// Kernel 1: per batch row b
//   wraw[n] = neighbor[b,n,:] . Ww_b
//   logits  = mask_c * (concept.Ww_a + wraw + bw) ; softmax over n
//   E[b,:]  = N*concept[b,:] + sum_n w[n]*neighbor[b,n,:]
//   maskS[b]= (sum stu[b,:] != 0)
// Reads the 134MB neighbor tensor exactly once (HBM-bound part).
// ---------------------------------------------------------------------------
__global__ __launch_bounds__(256) void k1_softmax_wavg(
    const float* __restrict__ stu, const float* __restrict__ con,
    const float* __restrict__ nbr, const float* __restrict__ Ww,
    const float* __restrict__ bw,  float* __restrict__ Eout,
    float* __restrict__ maskS)
{
    __shared__ float lds[NN * LROW];      // 64x132 floats = 33792 B
    __shared__ float wraw[NN];
    __shared__ float wv[NN];
    __shared__ float s_cdot, s_csum, s_ssum;

    const int b   = blockIdx.x;
    const int tid = threadIdx.x;
    const float* nb = nbr + (size_t)b * (NN * DD);

    // ---- Stage A: neighbor tile -> LDS (float4, fully coalesced) ----
    #pragma unroll
    for (int i = 0; i < 8; ++i) {
        int g   = tid + i * 256;          // float4 index 0..2047
        int row = g >> 5;                 // 32 float4 per row
        int c4  = g & 31;
        float4 v = ((const float4*)nb)[g];
        *(float4*)&lds[row * LROW + c4 * 4] = v;
    }

    // ---- Stage A2 (overlapped): cdot/csum on wave0, ssum on wave1 ----
    if (tid < 32) {
        const float4 cv = ((const float4*)(con + (size_t)b * DD))[tid];
        float pc = cv.x * Ww[tid*4+0] + cv.y * Ww[tid*4+1] +
                   cv.z * Ww[tid*4+2] + cv.w * Ww[tid*4+3];
        float ps = cv.x + cv.y + cv.z + cv.w;
        for (int o = 16; o > 0; o >>= 1) {
            pc += __shfl_xor(pc, o, 32);
            ps += __shfl_xor(ps, o, 32);
        }
        if (tid == 0) { s_cdot = pc; s_csum = ps; }
    } else if (tid < 64) {
        const int l = tid - 32;
        const float4 sv = ((const float4*)(stu + (size_t)b * DD))[l];
        float ps = sv.x + sv.y + sv.z + sv.w;
        for (int o = 16; o > 0; o >>= 1) ps += __shfl_xor(ps, o, 32);
        if (l == 0) s_ssum = ps;
    }
    __syncthreads();

    // ---- Stage B: wraw[n] = row_n . Ww_b  (4 lanes per row, stride-4
    //      interleave -> bank-conflict-free LDS reads) ----
    {
        const int n = tid >> 2;
        const int q = tid & 3;
        float p = 0.f;
        #pragma unroll
        for (int j = 0; j < 32; ++j) {
            const int d = q + 4 * j;
            p += lds[n * LROW + d] * Ww[DD + d];
        }
        p += __shfl_xor(p, 1, 32);
        p += __shfl_xor(p, 2, 32);
        if (q == 0) wraw[n] = p;
    }
    __syncthreads();

    // ---- Stage C: masked softmax over n (N=64, tiny) ----
    const float cdot = s_cdot;
    const float mc   = (s_csum != 0.0f) ? 1.0f : 0.0f;
    const float bw0  = bw[0];
    if (tid < NN) wraw[tid] = (cdot + wraw[tid] + bw0) * mc;  // logits
    __syncthreads();
    float m = -3.402823466e+38f;
    for (int n = 0; n < NN; ++n) m = fmaxf(m, wraw[n]);       // broadcast reads
    if (tid < NN) wv[tid] = __expf(wraw[tid] - m);
    __syncthreads();
    float se = 0.f;
    for (int n = 0; n < NN; ++n) se += wv[n];                 // broadcast reads
    const float inv_s = 1.0f / se;

    // ---- Stage D: E[b,d] = N*concept + sum_n w[n]*nbr[n,d] ----
    if (tid < DD) {
        float acc = 0.f;
        #pragma unroll 8
        for (int n = 0; n < NN; ++n) acc += wv[n] * lds[n * LROW + tid];
        const float wavg = acc * inv_s;
        Eout[(size_t)b * DD + tid] = 64.0f * con[(size_t)b * DD + tid] + wavg;
    }
    if (tid == 0) maskS[b] = (s_ssum != 0.0f) ? 1.0f : 0.0f;
}

// ---------------------------------------------------------------------------
// Kernel 2: y = 0.5*mask_s*( [65*stu | E](Bx256) @ Ws(256x128) + 65*bs )
// One block = 16 output rows, one wave per 16-col tile, K=256 via
// v_wmma_f32_16x16x4_f32 (fp32-exact). Two accumulators to pipeline.
// ---------------------------------------------------------------------------
__global__ __launch_bounds__(256) void k2_wmma_gemm(
    const float* __restrict__ stu, const float* __restrict__ E,
    const float* __restrict__ Ws,  const float* __restrict__ bs,
    const float* __restrict__ maskS, float* __restrict__ y)
{
    __shared__ float Alds[16 * ASTR];     // 16x260 floats = 16640 B
    const int tid = threadIdx.x;
    const int m0  = blockIdx.x * 16;

    // Stage A' = [65*stu | E] rows m0..m0+15 into LDS (float4, coalesced)
    #pragma unroll
    for (int i = 0; i < 4; ++i) {
        int g   = tid + i * 256;          // float4 slot 0..1023 (16 rows x 64)
        int row = g >> 6;
        int c4  = g & 63;
        int k   = c4 * 4;
        float4 v;
        if (k < DD) {
            v = ((const float4*)(stu + (size_t)(m0 + row) * DD))[c4];
            v.x *= 65.f; v.y *= 65.f; v.z *= 65.f; v.w *= 65.f;
        } else {
            v = ((const float4*)(E + (size_t)(m0 + row) * DD))[c4 - 32];
        }
        *(float4*)&Alds[row * ASTR + k] = v;
    }
    __syncthreads();

    const int lane = tid & 31;
    const int wid  = tid >> 5;            // wave 0..7 -> column tile
    const int n0   = wid * 16;
    const int r    = lane & 15;
    const int ks2  = (lane >> 4) << 1;    // 0 (lanes 0-15: K=0,1) or 2 (K=2,3)

    v8f c0 = {0.f,0.f,0.f,0.f,0.f,0.f,0.f,0.f};
    v8f c1 = {0.f,0.f,0.f,0.f,0.f,0.f,0.f,0.f};
    const float* Ap = &Alds[r * ASTR + ks2];
    const float* Bp = Ws + (size_t)ks2 * DD + n0 + r;

    #pragma unroll 4
    for (int k0 = 0; k0 < 256; k0 += 8) {
        // A operand: lane holds (M=r, K=k0+ks2) and (M=r, K=k0+ks2+1)
        v2f a0; { float2 t = *(const float2*)(Ap + k0);     a0.x = t.x; a0.y = t.y; }
        v2f b0; b0.x = Bp[(size_t)(k0    ) * DD]; b0.y = Bp[(size_t)(k0 + 1) * DD];
        c0 = __builtin_amdgcn_wmma_f32_16x16x4_f32(false, a0, false, b0,
                                                   (short)0, c0, false, false);
        v2f a1; { float2 t = *(const float2*)(Ap + k0 + 4); a1.x = t.x; a1.y = t.y; }
        v2f b1; b1.x = Bp[(size_t)(k0 + 4) * DD]; b1.y = Bp[(size_t)(k0 + 5) * DD];
        c1 = __builtin_amdgcn_wmma_f32_16x16x4_f32(false, a1, false, b1,
                                                   (short)0, c1, false, false);
    }

    // Epilogue per documented C layout: c[v] -> row = v + 8*(lane>>4), col = r
    const float bsn     = 65.0f * bs[n0 + r];
    const int   rowbase = m0 + ((lane >> 4) << 3);
    #pragma unroll
    for (int v = 0; v < 8; ++v) {
        const int row = rowbase + v;
        const float mk = maskS[row];
        y[(size_t)row * DD + n0 + r] = (c0[v] + c1[v] + bsn) * mk * 0.5f;
    }
}

// ---------------------------------------------------------------------------
extern "C" void kernel_launch(void* const* d_in, const int* in_sizes, int n_in,
                              void* d_out, int out_size, void* d_ws, size_t ws_size,
                              hipStream_t stream) {
    (void)in_sizes; (void)n_in; (void)out_size; (void)ws_size;
    const float* stu = (const float*)d_in[0];   // (B, D)
    const float* con = (const float*)d_in[1];   // (B, D)
    const float* nbr = (const float*)d_in[2];   // (B, N, D)
    const float* Ws  = (const float*)d_in[3];   // (2D, D)
    const float* bs  = (const float*)d_in[4];   // (D,)
    const float* Ww  = (const float*)d_in[5];   // (2D, 1)
    const float* bw  = (const float*)d_in[6];   // (1,)

    float* E     = (float*)d_ws;                // B*D floats = 2 MB
    float* maskS = E + (size_t)BATCH * DD;      // B floats
    float* yout  = (float*)d_out;               // (B, D)

    k1_softmax_wavg<<<BATCH, 256, 0, stream>>>(stu, con, nbr, Ww, bw, E, maskS);
    k2_wmma_gemm<<<BATCH / 16, 256, 0, stream>>>(stu, E, Ws, bs, maskS, yout);
}